// SpectralPooling_86165633893073
// MI455X (gfx1250) — compile-verified
//
#include <hip/hip_runtime.h>
#include <math.h>

// ---------------------------------------------------------------------------
// SpectralPooling on MI455X (gfx1250):
//   y = Re( IFFT3_32( pad32( crop24( Re( FFT3_48(x) ) ) ) ) ), 256 volumes.
// 6 batched small GEMMs (one per axis, forward then inverse) using
// V_WMMA_F32_16X16X4_F32 for full fp32 accuracy.
// Stage form: out[v,j,p,q] = sum_l T[j,l] * in[v,p,q,l]  (cyclic axis rotate)
// Twiddle (A) operands are hoisted into registers in a prolog; the unrolled
// steady-state loop is ds_load (B from padded LDS) + back-to-back WMMA only.
// ---------------------------------------------------------------------------

typedef float v2f __attribute__((ext_vector_type(2)));
typedef float v8f __attribute__((ext_vector_type(8)));

static __device__ __forceinline__ v8f wmma_f32_k4(v2f a, v2f b, v8f c) {
  // D = A(16x4) * B(4x16) + C, fp32.
  // (neg_a, A, neg_b, B, c_mod, C, reuse_a, reuse_b). f32 WMMA has no A/B neg.
  return __builtin_amdgcn_wmma_f32_16x16x4_f32(false, a, false, b,
                                               (short)0, c, false, false);
}

// VAR: 0 = real-in -> complex-out
//      1 = complex-in -> complex-out
//      2 = complex-in -> Re-only-out
// One workgroup handles one (volume, p) panel: full JxQ output tile set.
// blockDim.x = 32 * ceil(J/16) * ceil(Q/16)  (<= 192).
template <int L, int VAR>
__global__ void __launch_bounds__(192)
spectral_stage(const float* __restrict__ in_re, const float* __restrict__ in_im,
               float* __restrict__ out_re, float* __restrict__ out_im,
               const float* __restrict__ T_re, const float* __restrict__ T_im,
               int P, int Q, int J)
{
  constexpr int KS    = L / 4;     // K-steps
  constexpr int rowst = L + 1;     // odd LDS row stride -> conflict-free

  // LDS panel: rows = Q rounded to 16 (zero-padded). Worst case 2*48*49 floats.
  __shared__ float X[2 * 48 * 49];

  const int b = blockIdx.x;            // v*P + p
  const int v = b / P;
  const int p = b - v * P;
  const int QT16  = (Q + 15) & ~15;
  const int imoff = QT16 * rowst;
  const int tid  = threadIdx.x;
  const int nthr = blockDim.x;

  // Wave -> (j-tile, q-tile)
  const int wave = tid >> 5;
  const int lane = tid & 31;
  const int JT = (J + 15) >> 4;
  const int jt = wave % JT;
  const int qt = wave / JT;
  const int jb = jt << 4, qb = qt << 4;

  // ISA operand layouts (wave32):
  //  A 16x4 : lane m=lane&15 holds T[jb+m, k + 2*(lane>=16) + {0,1}]
  //  B 4x16 : lane n=lane&15 holds X[qb+n, k + 2*(lane>=16) + {0,1}]
  //  C 16x16: elem r of lane -> row jb + r + 8*(lane>=16), col qb + (lane&15)
  const int m  = lane & 15;
  const int kh = (lane >> 4) << 1;   // 0 or 2
  int trow = jb + m; if (trow >= J) trow = J - 1;   // clamp (stores masked)
  const float* Tr = T_re + (size_t)trow * L;
  const float* Ti = T_im + (size_t)trow * L;

  // ---- Prolog: pull all twiddle operands into registers (loop-invariant) ---
  v2f ar[KS], ai[KS], an[KS];
#pragma unroll
  for (int s = 0; s < KS; ++s) {
    const int k = 4 * s + kh;
    ar[s].x = Tr[k]; ar[s].y = Tr[k + 1];
    v2f t;  t.x  = Ti[k]; t.y  = Ti[k + 1];
    if (VAR != 2) ai[s] = t;       // +Ti for imag accumulation
    if (VAR != 0) an[s] = -t;      // -Ti for real accumulation (no A-neg in HW)
  }

  // ---- Cooperative, coalesced panel load into LDS (pad rows zero-filled) ---
  const float* src_re = in_re + (size_t)b * Q * L;
  const float* src_im = (VAR != 0) ? in_im + (size_t)b * Q * L : nullptr;
  for (int idx = tid; idx < QT16 * L; idx += nthr) {
    int q = idx / L;
    int l = idx - q * L;
    X[q * rowst + l] = (q < Q) ? src_re[q * L + l] : 0.0f;
    if (VAR != 0)
      X[imoff + q * rowst + l] = (q < Q) ? src_im[q * L + l] : 0.0f;
  }
  __syncthreads();

  const float* Xr = X + (qb + m) * rowst;
  const float* Xi = Xr + imoff;

  v8f acc_re = {0, 0, 0, 0, 0, 0, 0, 0};
  v8f acc_im = {0, 0, 0, 0, 0, 0, 0, 0};

  // ---- Steady state: LDS B-operand loads + WMMA only -----------------------
#pragma unroll
  for (int s = 0; s < KS; ++s) {
    const int k = 4 * s + kh;
    v2f br; br.x = Xr[k]; br.y = Xr[k + 1];
    if (VAR == 0) {
      acc_re = wmma_f32_k4(ar[s], br, acc_re);
      acc_im = wmma_f32_k4(ai[s], br, acc_im);
    } else {
      v2f bi; bi.x = Xi[k]; bi.y = Xi[k + 1];
      acc_re = wmma_f32_k4(ar[s], br, acc_re);
      acc_re = wmma_f32_k4(an[s], bi, acc_re);   // re -= Ti*Xi
      if (VAR == 1) {
        acc_im = wmma_f32_k4(ai[s], br, acc_im);
        acc_im = wmma_f32_k4(ar[s], bi, acc_im);
      }
    }
  }

  // ---- Store (mask j>=J for J=24 tiles and q>=Q for Q=24 tiles) ------------
  const int q = qb + m;
  const int jrow0 = jb + ((lane >> 4) << 3);
  if (q < Q) {
#pragma unroll
    for (int r = 0; r < 8; ++r) {
      int j = jrow0 + r;
      if (j < J) {
        size_t o = (((size_t)v * J + j) * P + p) * Q + q;
        out_re[o] = acc_re[r];
        if (VAR == 1 || VAR == 0) out_im[o] = acc_im[r];
      }
    }
  }
}

// Twiddle matrices:
//  A (24x48): forward-FFT rows k=0..23 (crop fused), scale 1/sqrt(48)
//  B (32x24): inverse-FFT cols k=0..23 (pad fused),  scale 1/sqrt(32)
__global__ void fill_twiddles(float* __restrict__ Are, float* __restrict__ Aim,
                              float* __restrict__ Bre, float* __restrict__ Bim)
{
  int t = blockIdx.x * blockDim.x + threadIdx.x;
  const float PI2 = 6.283185307179586f;
  if (t < 24 * 48) {
    int k = t / 48, n = t - k * 48;
    int r = (k * n) % 48;                       // exact phase reduction
    float ang = -PI2 * (float)r * (1.0f / 48.0f);
    const float s = 0.14433756729740643f;       // 1/sqrt(48)
    Are[t] = cosf(ang) * s;
    Aim[t] = sinf(ang) * s;
  }
  if (t < 32 * 24) {
    int mm = t / 24, k = t - mm * 24;
    int r = (mm * k) % 32;
    float ang = PI2 * (float)r * (1.0f / 32.0f);
    const float s = 0.17677669529663687f;       // 1/sqrt(32)
    Bre[t] = cosf(ang) * s;
    Bim[t] = sinf(ang) * s;
  }
}

extern "C" void kernel_launch(void* const* d_in, const int* in_sizes, int n_in,
                              void* d_out, int out_size, void* d_ws, size_t ws_size,
                              hipStream_t stream)
{
  const float* x = (const float*)d_in[0];
  float* out = (float*)d_out;
  float* ws  = (float*)d_ws;

  const int V = in_sizes[0] / (48 * 48 * 48);   // 256 batched volumes

  // Workspace layout (floats):
  float* Are = ws;                    // 24*48
  float* Aim = ws + 1152;             // 24*48
  float* Bre = ws + 2304;             // 32*24
  float* Bim = ws + 3072;             // 32*24
  float* bufA = ws + 4096;            // 2 * V*24*48*48 floats (113 MB)
  float* bufB = bufA + 2 * (size_t)V * 24 * 48 * 48;  // 2 * V*24*24*48 floats

  const size_t halfA1 = (size_t)V * 24 * 48 * 48;   // stage1 plane
  const size_t halfB2 = (size_t)V * 24 * 24 * 48;   // stage2 plane
  const size_t halfB4 = (size_t)V * 32 * 24 * 24;   // stage4 plane
  const size_t halfA5 = (size_t)V * 32 * 32 * 24;   // stage5 plane

  fill_twiddles<<<dim3(5), dim3(256), 0, stream>>>(Are, Aim, Bre, Bim);

  // S1: (V,48,48,48) real --A(n3)--> (V,24,48,48) complex   (2x3 tiles = 6 waves)
  spectral_stage<48, 0><<<dim3(V * 48), dim3(192), 0, stream>>>(
      x, nullptr, bufA, bufA + halfA1, Are, Aim, /*P=*/48, /*Q=*/48, /*J=*/24);

  // S2: (V,24,48,48) --A(n2)--> (V,24,24,48) complex
  spectral_stage<48, 1><<<dim3(V * 24), dim3(192), 0, stream>>>(
      bufA, bufA + halfA1, bufB, bufB + halfB2, Are, Aim, 24, 48, 24);

  // S3: (V,24,24,48) --A(n1)--> Re -> (V,24,24,24) real     (2x2 tiles = 4 waves)
  spectral_stage<48, 2><<<dim3(V * 24), dim3(128), 0, stream>>>(
      bufB, bufB + halfB2, bufA, nullptr, Are, Aim, 24, 24, 24);

  // S4: (V,24,24,24) real --B(k3)--> (V,32,24,24) complex
  spectral_stage<24, 0><<<dim3(V * 24), dim3(128), 0, stream>>>(
      bufA, nullptr, bufB, bufB + halfB4, Bre, Bim, 24, 24, 32);

  // S5: (V,32,24,24) --B(k2)--> (V,32,32,24) complex
  spectral_stage<24, 1><<<dim3(V * 32), dim3(128), 0, stream>>>(
      bufB, bufB + halfB4, bufA, bufA + halfA5, Bre, Bim, 32, 24, 32);

  // S6: (V,32,32,24) --B(k1)--> Re -> (V,32,32,32) real = d_out
  spectral_stage<24, 2><<<dim3(V * 32), dim3(128), 0, stream>>>(
      bufA, bufA + halfA5, out, nullptr, Bre, Bim, 32, 32, 32);
}